// ModelSGLang_60533269069825
// MI455X (gfx1250) — compile-verified
//
#include <hip/hip_runtime.h>
#include <stdint.h>

// ---------------------------------------------------------------------------
// KV-cache scatter write for MI455X (gfx1250).
//   out[0 .. 500000*576)          = kv_buffer (bulk streaming copy, ~2.3 GB traffic)
//   out[loc[t]*576 + 0   .. 512)  = cache_k_nope[t]
//   out[loc[t]*576 + 512 .. 576)  = cache_k_rope[t]
// Pure bandwidth problem (~105 us at 23.3 TB/s): use CDNA5 async global<->LDS
// DMA (ASYNCcnt) for the bulk copy and non-temporal B128 ops for the scatter.
// ---------------------------------------------------------------------------

typedef __attribute__((ext_vector_type(4))) float v4f;
typedef __attribute__((ext_vector_type(4))) int   v4i;

// Builtin parameter types: int4* in global (AS1) and LDS (AS3) address spaces.
typedef __attribute__((address_space(1))) v4i g_v4i;
typedef __attribute__((address_space(3))) v4i l_v4i;

#define COPY_BLOCKS   4096
#define COPY_THREADS  256
#define COPY_UNROLL   4

// gfx12 CPol: TH in bits [2:0]; TH=1 -> non-temporal (stream, don't rinse L2).
#define CPOL_NT 1

#if __has_builtin(__builtin_amdgcn_global_load_async_to_lds_b128) && \
    __has_builtin(__builtin_amdgcn_global_store_async_from_lds_b128)
#define HAVE_ASYNC_LDS 1
#else
#define HAVE_ASYNC_LDS 0
#endif

#if __has_builtin(__builtin_amdgcn_s_wait_asynccnt)
#define WAIT_ASYNC0() __builtin_amdgcn_s_wait_asynccnt(0)
#else
#define WAIT_ASYNC0() asm volatile("s_wait_asynccnt 0x0" ::: "memory")
#endif

// ---------------------------------------------------------------------------
// Bulk copy: src -> dst, nchunks 16-byte chunks.
// Async path: per phase, each lane DMAs 4 x 16B global->LDS, waits ASYNCcnt,
// then DMAs the same 4 x 16B LDS->global. No VGPR data movement at all.
// ---------------------------------------------------------------------------
__global__ __launch_bounds__(COPY_THREADS)
void kv_bulk_copy(const v4f* __restrict__ src, v4f* __restrict__ dst,
                  long long nchunks) {
  const int tid = threadIdx.x;
#if HAVE_ASYNC_LDS
  __shared__ __align__(16) v4f stage[COPY_UNROLL][COPY_THREADS];
  const long long stride = (long long)gridDim.x * (COPY_THREADS * COPY_UNROLL);
  for (long long it = (long long)blockIdx.x * (COPY_THREADS * COPY_UNROLL);
       it < nchunks; it += stride) {
#pragma unroll
    for (int u = 0; u < COPY_UNROLL; ++u) {
      long long idx = it + (long long)u * COPY_THREADS + tid;
      if (idx < nchunks) {
        __builtin_amdgcn_global_load_async_to_lds_b128(
            (g_v4i*)(src + idx), (l_v4i*)&stage[u][tid],
            /*imm offset*/ 0, /*cpol*/ CPOL_NT);
      }
    }
    WAIT_ASYNC0();  // loads complete -> LDS data valid
#pragma unroll
    for (int u = 0; u < COPY_UNROLL; ++u) {
      long long idx = it + (long long)u * COPY_THREADS + tid;
      if (idx < nchunks) {
        __builtin_amdgcn_global_store_async_from_lds_b128(
            (g_v4i*)(dst + idx), (l_v4i*)&stage[u][tid],
            /*imm offset*/ 0, /*cpol*/ CPOL_NT);
      }
    }
    WAIT_ASYNC0();  // stores complete -> LDS staging reusable
  }
#else
  // Fallback: plain streaming copy with non-temporal B128 accesses.
  const long long stride = (long long)gridDim.x * (COPY_THREADS * COPY_UNROLL);
  for (long long it = (long long)blockIdx.x * (COPY_THREADS * COPY_UNROLL);
       it < nchunks; it += stride) {
#pragma unroll
    for (int u = 0; u < COPY_UNROLL; ++u) {
      long long idx = it + (long long)u * COPY_THREADS + tid;
      if (idx < nchunks) {
        v4f v = __builtin_nontemporal_load(src + idx);
        __builtin_nontemporal_store(v, dst + idx);
      }
    }
  }
#endif
}

// ---------------------------------------------------------------------------
// Scatter: one 128-thread block per token. Rows are 576 floats (2304 B,
// 16B-aligned). Lanes 0..127 move nope (512 f32 = 128 x b128),
// lanes 0..15 additionally move rope (64 f32 = 16 x b128).
// ---------------------------------------------------------------------------
__global__ __launch_bounds__(128)
void kv_scatter(const long long* __restrict__ loc,
                const v4f* __restrict__ nope,   // [T * 128] chunks
                const v4f* __restrict__ rope,   // [T * 16] chunks
                float* __restrict__ out) {
  const int t   = blockIdx.x;
  const int tid = threadIdx.x;
  const long long slot = loc[t];
  v4f* dst = (v4f*)(out + slot * 576LL);

  v4f v = __builtin_nontemporal_load(nope + (long long)t * 128 + tid);
  __builtin_nontemporal_store(v, dst + tid);

  if (tid < 16) {
    v4f r = __builtin_nontemporal_load(rope + (long long)t * 16 + tid);
    __builtin_nontemporal_store(r, dst + 128 + tid);
  }
}

// ---------------------------------------------------------------------------
extern "C" void kernel_launch(void* const* d_in, const int* in_sizes, int n_in,
                              void* d_out, int out_size, void* d_ws, size_t ws_size,
                              hipStream_t stream) {
  const float*     kv   = (const float*)d_in[0];       // [500000, 1, 576] f32 (zeros)
  const long long* loc  = (const long long*)d_in[1];   // [T] int64, unique
  const float*     nope = (const float*)d_in[2];       // [T, 1, 512] f32
  const float*     rope = (const float*)d_in[3];       // [T, 1, 64] f32
  float*           out  = (float*)d_out;               // [500000, 1, 576] f32
  (void)n_in; (void)d_ws; (void)ws_size;

  const long long nchunks = (long long)out_size / 4;   // 16-byte chunks
  const int T = in_sizes[1];                           // number of tokens

  // 1) materialize kv_buffer into d_out (streaming, async LDS DMA path)
  kv_bulk_copy<<<COPY_BLOCKS, COPY_THREADS, 0, stream>>>(
      (const v4f*)kv, (v4f*)out, nchunks);

  // 2) scatter the 576-float rows into their slots
  kv_scatter<<<T, 128, 0, stream>>>(loc, (const v4f*)nope, (const v4f*)rope, out);
}